// DiffusionModel2d_34273839022159
// MI455X (gfx1250) — compile-verified
//
#include <hip/hip_runtime.h>
#include <stdint.h>

typedef uint32_t u32;
typedef uint64_t u64;
typedef u32 v4u __attribute__((ext_vector_type(4)));
typedef int  v8i __attribute__((ext_vector_type(8)));
typedef int  v4i __attribute__((ext_vector_type(4)));

#define MG      256
#define NCELL   (MG * MG)          // 65536 cells per batch
#define HINV2F  65536.0f           // 1/h^2, h = 1/256
#define NT      1024               // threads per CG block (32 wave32 waves)
#define CPT     (NCELL / NT)       // 64 cells per thread
#define NWAVE   (NT / 32)          // 32 waves
#define NITER   300                // CG maxiter (fixed, deterministic)
#define TJ      32                 // coefficient rows per precompute tile
#define AW      (MG + 1)           // 257: alpha vertex grid width

// d_ws layout per batch (floats):
//   [0 .. 4*NCELL)  : float4 {diag, K_left, K_right, K_bottom} (pre-scaled by 1/h^2)
//   [4N .. 5N)      : K_top (pre-scaled)
//   [5N .. 6N)      : r
//   [6N .. 7N)      : Ap
#define WS_PER_BATCH ((size_t)7 * NCELL)

// ---------------------------------------------------------------------------
// Workgroup-wide sum reduction (wave32 shuffle + LDS across 32 waves).
// ---------------------------------------------------------------------------
__device__ __forceinline__ float block_reduce(float v, float* red) {
#pragma unroll
    for (int off = 16; off > 0; off >>= 1) v += __shfl_down(v, off, 32);
    const int lane = threadIdx.x & 31;
    const int w    = threadIdx.x >> 5;
    if (lane == 0) red[w] = v;
    __syncthreads();
    if (w == 0) {
        float s = red[lane];
#pragma unroll
        for (int off = 16; off > 0; off >>= 1) s += __shfl_down(s, off, 32);
        if (lane == 0) red[NWAVE] = s;
    }
    __syncthreads();
    float r = red[NWAVE];
    __syncthreads();
    return r;
}

// ---------------------------------------------------------------------------
// Kernel 1: build stencil coefficients from alpha.
// One block per (batch, 32-row tile). The (TJ+1) x 257 alpha tile is DMA'd
// into LDS by the Tensor Data Mover (tensor_load_to_lds), then 256 threads
// compute exp-averaged edge coefficients and store them pre-scaled by 1/h^2.
// ---------------------------------------------------------------------------
__global__ __launch_bounds__(256) void coeff_kernel(const float* __restrict__ alpha,
                                                    float* __restrict__ ws) {
    __shared__ float At[(TJ + 1) * AW];   // 33 x 257 alpha tile (~33 KB)

    const int b  = blockIdx.x >> 3;       // batch
    const int jt = blockIdx.x & 7;        // row-tile within batch
    const int j0 = jt * TJ;

    // ---- TDM: async 2D tile load alpha[b, j0 .. j0+TJ, 0 .. 256] -> LDS ----
    {
        const float* gsrc = alpha + (size_t)b * AW * AW + (size_t)j0 * AW;
        const u64 ga  = (u64)(uintptr_t)gsrc;
        const u32 lds = (u32)(uintptr_t)(&At[0]);   // low 32 bits = LDS offset

        // D# group 0: count=1 | lds_addr | global_addr(57b) | type=2
        v4u g0 = { 1u, lds, (u32)ga,
                   (u32)((ga >> 32) & 0x1FFFFFFu) | 0x80000000u };

        // D# group 1: data_size=4B, tensor 257x257, tile 257 x (TJ+1),
        // dim0 stride 257, dim1 stride 257*257 (element units).
        const u32 TD0 = AW, TD1 = AW, TL0 = AW, TL1 = TJ + 1;
        const u64 S0 = AW, S1 = (u64)AW * AW;
        v8i g1 = { (int)(2u << 16),                                   // data_size=2 (4B)
                   (int)((TD0 & 0xFFFFu) << 16),                      // tensor_dim0 lo16
                   (int)((TD0 >> 16) | ((TD1 & 0xFFFFu) << 16)),      // td0 hi | td1 lo
                   (int)((TD1 >> 16) | (TL0 << 16)),                  // td1 hi | tile_dim0
                   (int)TL1,                                          // tile_dim1 (tile_dim2=0)
                   (int)(u32)(S0 & 0xFFFFFFFFu),                      // stride0 lo32
                   (int)((u32)((S0 >> 32) & 0xFFFFu) |
                         ((u32)(S1 & 0xFFFFu) << 16)),                // stride0 hi | stride1 lo
                   (int)(u32)((S1 >> 16) & 0xFFFFFFFFu) };            // stride1 hi
        v4i g2 = {0, 0, 0, 0};
        v4i g3 = {0, 0, 0, 0};

        if (threadIdx.x == 0) {
#if defined(__clang_major__) && (__clang_major__ >= 23)
            v8i gx = {0, 0, 0, 0, 0, 0, 0, 0};
            __builtin_amdgcn_tensor_load_to_lds(g0, g1, g2, g3, gx, 0);
#else
            __builtin_amdgcn_tensor_load_to_lds(g0, g1, g2, g3, 0);
#endif
        }
        __builtin_amdgcn_s_wait_tensorcnt(0);
    }
    __syncthreads();

    float* base = ws + (size_t)b * WS_PER_BATCH;
    float4* C4  = (float4*)base;
    float* KT   = base + 4 * (size_t)NCELL;

#pragma unroll 4
    for (int i = 0; i < (TJ * MG) / 256; ++i) {
        const int l  = threadIdx.x + i * 256;     // local cell in tile
        const int jj = l >> 8;                    // local row (0..TJ-1)
        const int k  = l & (MG - 1);              // column
        const int j  = j0 + jj;                   // global row

        const float a00 = At[jj * AW + k];
        const float a01 = At[jj * AW + k + 1];
        const float a10 = At[(jj + 1) * AW + k];
        const float a11 = At[(jj + 1) * AW + k + 1];

        const float kl = expf(0.5f * (a00 + a01));
        const float kr = (j < MG - 1) ? expf(0.5f * (a10 + a11)) : 0.f;
        const float kb = (k > 0)      ? expf(0.5f * (a00 + a10)) : 0.f;
        const float kt = (k < MG - 1) ? expf(0.5f * (a01 + a11)) : 0.f;
        const float dg = kl + kr + kb + kt + ((j == 0) ? kl : 0.f);

        const int idx = j * MG + k;
        float4 cc;
        cc.x = HINV2F * dg;
        cc.y = HINV2F * kl;
        cc.z = HINV2F * kr;
        cc.w = HINV2F * kb;
        C4[idx] = cc;
        KT[idx] = HINV2F * kt;
    }
}

// ---------------------------------------------------------------------------
// Kernel 2: persistent CG, one block per batch. p lives in LDS (256 KB);
// coefficients / r / Ap stream from L2; u accumulates directly into d_out.
// ---------------------------------------------------------------------------
__global__ __launch_bounds__(NT) void cg_kernel(const float* __restrict__ f,
                                                float* __restrict__ ws,
                                                float* __restrict__ out) {
    extern __shared__ float P[];          // 65536 floats: search direction p
    __shared__ float red[NWAVE + 1];

    const int b   = blockIdx.x;
    const int tid = threadIdx.x;

    float* base = ws + (size_t)b * WS_PER_BATCH;
    const float4* __restrict__ C4 = (const float4*)base;
    const float*  __restrict__ KT = base + 4 * (size_t)NCELL;
    float* R  = base + 5 * (size_t)NCELL;
    float* AP = base + 6 * (size_t)NCELL;
    float* U  = out + (size_t)b * NCELL;

    // init: u = 0, r = p = f, rr = <f,f>
    float loc = 0.f;
#pragma unroll 4
    for (int c = 0; c < CPT; ++c) {
        const int idx = tid + c * NT;
        const float fv = f[idx];
        P[idx] = fv;
        R[idx] = fv;
        U[idx] = 0.f;
        loc += fv * fv;
    }
    float rr = block_reduce(loc, red);

    for (int it = 0; it < NITER; ++it) {
        // ---- Ap = A p (stencil from LDS), pAp = <p, Ap> ----
        loc = 0.f;
#pragma unroll 2
        for (int c = 0; c < CPT; ++c) {
            const int idx = tid + c * NT;
            const int jj = idx >> 8;
            const int kk = idx & (MG - 1);
            const float4 cc = C4[idx];
            const float ct = KT[idx];
            const float pc = P[idx];
            const float pm = (jj > 0)      ? P[idx - MG] : 0.f;  // p[j-1,k]
            const float pp = (jj < MG - 1) ? P[idx + MG] : 0.f;  // p[j+1,k]
            const float pl = (kk > 0)      ? P[idx - 1]  : 0.f;  // p[j,k-1]
            const float pr = (kk < MG - 1) ? P[idx + 1]  : 0.f;  // p[j,k+1]
            const float ap = cc.x * pc - cc.y * pm - cc.z * pp - cc.w * pl - ct * pr;
            AP[idx] = ap;
            loc += ap * pc;
        }
        const float pAp = block_reduce(loc, red);
        const float a = (pAp != 0.f) ? (rr / pAp) : 0.f;

        // ---- u += a p ; r -= a Ap ; rr_new = <r,r> ----
        loc = 0.f;
#pragma unroll 2
        for (int c = 0; c < CPT; ++c) {
            const int idx = tid + c * NT;
            const float pv = P[idx];
            U[idx] += a * pv;
            const float rv = R[idx] - a * AP[idx];
            R[idx] = rv;
            loc += rv * rv;
        }
        const float rrn = block_reduce(loc, red);
        const float bta = (rr != 0.f) ? (rrn / rr) : 0.f;
        rr = rrn;

        // ---- p = r + beta p ----
#pragma unroll 4
        for (int c = 0; c < CPT; ++c) {
            const int idx = tid + c * NT;
            P[idx] = R[idx] + bta * P[idx];
        }
        __syncthreads();
    }
}

// ---------------------------------------------------------------------------
extern "C" void kernel_launch(void* const* d_in, const int* in_sizes, int n_in,
                              void* d_out, int out_size, void* d_ws, size_t ws_size,
                              hipStream_t stream) {
    (void)n_in; (void)out_size; (void)ws_size;
    const float* alpha = (const float*)d_in[0];   // (B, 257, 257) fp32
    const float* f     = (const float*)d_in[1];   // (256, 256) fp32
    float* out = (float*)d_out;                   // (B, 256, 256) fp32
    float* ws  = (float*)d_ws;

    const int B = in_sizes[0] / (AW * AW);        // 64

    hipFuncSetAttribute(reinterpret_cast<const void*>(cg_kernel),
                        hipFuncAttributeMaxDynamicSharedMemorySize,
                        NCELL * (int)sizeof(float));

    coeff_kernel<<<dim3(B * (MG / TJ)), 256, 0, stream>>>(alpha, ws);
    cg_kernel<<<dim3(B), NT, NCELL * sizeof(float), stream>>>(f, ws, out);
}